// GraphAttentionLayer_11768210391527
// MI455X (gfx1250) — compile-verified
//
#include <hip/hip_runtime.h>
#include <hip/hip_bf16.h>

// ---------------------------------------------------------------------------
// GraphAttentionLayer for MI455X (gfx1250), fp32 end-to-end.
//   N=768, IN_DIM=128, OUT_DIM=128, H=4, ATTN_WIDTH=256
// GEMMs: V_WMMA_F32_16X16X4_F32, one 16x64 C strip per wave (A reuse x4).
// Attention tile staging: GLOBAL_LOAD_ASYNC_TO_LDS_B128 + s_wait_asynccnt.
// ---------------------------------------------------------------------------

typedef float v2f __attribute__((ext_vector_type(2)));
typedef float v8f __attribute__((ext_vector_type(8)));

#define GN 768
#define GD 128
#define GH 4
#define GAW 256
#define NEG_SLOPE 0.01f

__device__ __forceinline__ float leaky(float x) {
    return fmaxf(x, NEG_SLOPE * x);   // leaky_relu(x, 0.01)
}

// -------- CDNA5 async global->LDS copy (ASYNCcnt-tracked, §15.18.3) --------
__device__ __forceinline__ void async_copy_b128(unsigned lds_off, const void* gaddr) {
    asm volatile("global_load_async_to_lds_b128 %0, %1, off"
                 :: "v"(lds_off), "v"(gaddr)
                 : "memory");
}
__device__ __forceinline__ void wait_async0() {
    asm volatile("s_wait_asynccnt 0x0" ::: "memory");
}

// ---------------------------------------------------------------------------
// fp32 WMMA GEMM:  C[m,n] = sum_k A[m,k] * B(k,n)
//   A row-major (contiguous k, row stride lda)
//   B(k,n) at B[k*sbk + n*sbn];  SBK1 => sbk==1 at compile time (b64 loads)
//   C[m,n] at C[m*ldc + n*scn]
// One wave computes a 16x(16*NT) strip; K stepped by 4.
// M%16==0, Nn%(16*NT)==0, K%4==0 for every call below.
// ---------------------------------------------------------------------------
template <bool SBK1, int NT>
__global__ void wmma_gemm_f32(const float* __restrict__ A,
                              const float* __restrict__ B,
                              float* __restrict__ C,
                              int M, int Nn, int K,
                              int lda, int sbk, int sbn, int ldc, int scn) {
    const int wavesPerBlock = blockDim.x >> 5;
    const int wave = threadIdx.x >> 5;
    const int lane = threadIdx.x & 31;
    const int groupsN = Nn / (16 * NT);
    const int totalTiles = (M >> 4) * groupsN;
    const int tile = blockIdx.x * wavesPerBlock + wave;
    if (tile >= totalTiles) return;          // wave-uniform: EXEC stays all-1s

    const int tm = tile / groupsN;
    const int tg = tile % groupsN;
    const int m0 = tm << 4;
    const int n0 = tg * (16 * NT);
    const int half = lane >> 4;              // 0: K={0,1}, 1: K={2,3}
    const int l = lane & 15;

    // A lane layout (16x4 f32): lane l -> M=l%16, K = half*2 + vgpr
    const float* aptr = A + (size_t)(m0 + l) * lda + (half << 1);
    // B lane layout (4x16 f32), mirrored: N = l%16, K = half*2 + vgpr
    const float* bptr[NT];
    #pragma unroll
    for (int tn = 0; tn < NT; ++tn) {
        bptr[tn] = B + (size_t)(n0 + tn * 16 + l) * sbn + (size_t)(half << 1) * sbk;
    }

    v8f acc[NT];
    #pragma unroll
    for (int tn = 0; tn < NT; ++tn) acc[tn] = (v8f){};

    for (int k0 = 0; k0 < K; k0 += 4) {
        if ((k0 & 63) == 0) {
            __builtin_prefetch(aptr + 64, 0, 0);   // global_prefetch_b8
        }
        v2f av;
        av[0] = aptr[0];
        av[1] = aptr[1];
        #pragma unroll
        for (int tn = 0; tn < NT; ++tn) {
            v2f bv;
            if (SBK1) {                       // contiguous: single b64 load
                bv[0] = bptr[tn][0];
                bv[1] = bptr[tn][1];
                bptr[tn] += 4;
            } else {
                bv[0] = bptr[tn][0];
                bv[1] = bptr[tn][sbk];
                bptr[tn] += (size_t)4 * sbk;
            }
            acc[tn] = __builtin_amdgcn_wmma_f32_16x16x4_f32(
                /*neg_a=*/false, av, /*neg_b=*/false, bv,
                /*c_mod=*/(short)0, acc[tn], /*reuse_a=*/false, /*reuse_b=*/false);
        }
        aptr += 4;
    }

    // C/D layout: VGPR v -> row m0 + v + half*8, col n0 + tn*16 + l
    #pragma unroll
    for (int tn = 0; tn < NT; ++tn) {
        float* cbase = C + (size_t)(n0 + tn * 16 + l) * scn;
        #pragma unroll
        for (int v = 0; v < 8; ++v) {
            cbase[(size_t)(m0 + v + (half << 3)) * ldc] = acc[tn][v];
        }
    }
}

// ---------------------------------------------------------------------------
// Pairwise attention logits.
//   hid[i,j,:] = leaky(Am[i,:] + Bm[j,:] + b1)
//   S[h][i][j] = leaky(dot(hid, w2[h,:]) + b2[h])   (planar per-head layout)
// Block = 256 threads handles a 16x16 (i,j) tile; slabs staged in LDS via
// async global->LDS b128 copies (the 16x256 slabs are contiguous 16KB).
// ---------------------------------------------------------------------------
__global__ void attn_logits_kernel(const float* __restrict__ Am,
                                   const float* __restrict__ Bm,
                                   const float* __restrict__ b1,
                                   const float* __restrict__ w2,
                                   const float* __restrict__ b2,
                                   float* __restrict__ S) {
    __shared__ float As[16][GAW];
    __shared__ float Bs[16][GAW];
    __shared__ float w2s[GH][GAW];
    __shared__ float b1s[GAW];
    __shared__ float b2s[GH];

    const int t = threadIdx.x;           // 0..255
    const int i0 = blockIdx.y << 4;
    const int j0 = blockIdx.x << 4;

    // Async stage: As/Bs are contiguous 16KB blocks (row stride == GAW).
    const unsigned asBase = (unsigned)(size_t)&As[0][0];
    const unsigned bsBase = (unsigned)(size_t)&Bs[0][0];
    const unsigned w2Base = (unsigned)(size_t)&w2s[0][0];
    const unsigned b1Base = (unsigned)(size_t)&b1s[0];
    const char* gA = (const char*)(Am + (size_t)i0 * GAW);
    const char* gB = (const char*)(Bm + (size_t)j0 * GAW);
    #pragma unroll
    for (int r = 0; r < 4; ++r) {
        const unsigned off = (unsigned)(t + (r << 8)) << 4;   // 16B per lane
        async_copy_b128(asBase + off, gA + off);
        async_copy_b128(bsBase + off, gB + off);
    }
    async_copy_b128(w2Base + ((unsigned)t << 4), (const char*)w2 + (t << 4));
    if (t < 64) {
        async_copy_b128(b1Base + ((unsigned)t << 4), (const char*)b1 + (t << 4));
    }
    if (t < GH) b2s[t] = b2[t];
    wait_async0();
    __syncthreads();

    // Fold b1 into the A slab once.
    for (int idx = t; idx < 16 * GAW; idx += 256) {
        As[idx >> 8][idx & (GAW - 1)] += b1s[idx & (GAW - 1)];
    }
    __syncthreads();

    const int i = t >> 4;
    const int j = t & 15;

    float acc0 = 0.f, acc1 = 0.f, acc2 = 0.f, acc3 = 0.f;
    #pragma unroll 4
    for (int k = 0; k < GAW; ++k) {
        const float x = As[i][k] + Bs[j][k];
        const float hid = fmaxf(x, NEG_SLOPE * x);
        acc0 = fmaf(hid, w2s[0][k], acc0);
        acc1 = fmaf(hid, w2s[1][k], acc1);
        acc2 = fmaf(hid, w2s[2][k], acc2);
        acc3 = fmaf(hid, w2s[3][k], acc3);
    }

    const size_t off = (size_t)(i0 + i) * GN + (j0 + j);
    const size_t NP = (size_t)GN * GN;
    S[0 * NP + off] = leaky(acc0 + b2s[0]);
    S[1 * NP + off] = leaky(acc1 + b2s[1]);
    S[2 * NP + off] = leaky(acc2 + b2s[2]);
    S[3 * NP + off] = leaky(acc3 + b2s[3]);
}

// ---------------------------------------------------------------------------
// Masked softmax over the neighbor axis, in place on planar S[h][n][:].
// Every row has a self-loop, so the row-wise masked max is exact (the global
// "initial" scalar in the reference is provably never selected).
// ---------------------------------------------------------------------------
__global__ void masked_softmax_kernel(float* __restrict__ S,
                                      const float* __restrict__ adj) {
    const int n = blockIdx.x;
    const int h = blockIdx.y;
    float* row = S + ((size_t)h * GN + n) * GN;
    const float* arow = adj + (size_t)n * GN;   // adj trailing dim is 1

    __shared__ float red[256];
    const int t = threadIdx.x;

    float mx = -3.4e38f;
    for (int j = t; j < GN; j += 256) {
        if (arow[j] > 0.f) mx = fmaxf(mx, row[j]);
    }
    red[t] = mx;
    __syncthreads();
    for (int s = 128; s > 0; s >>= 1) {
        if (t < s) red[t] = fmaxf(red[t], red[t + s]);
        __syncthreads();
    }
    mx = red[0];
    __syncthreads();

    float sum = 0.f;
    for (int j = t; j < GN; j += 256) {
        const float e = (arow[j] > 0.f) ? __expf(row[j] - mx) : 0.f;
        row[j] = e;
        sum += e;
    }
    red[t] = sum;
    __syncthreads();
    for (int s = 128; s > 0; s >>= 1) {
        if (t < s) red[t] += red[t + s];
        __syncthreads();
    }
    const float inv = 1.f / red[0];

    for (int j = t; j < GN; j += 256) row[j] *= inv;
}

// ---------------------------------------------------------------------------
// Host-side orchestration
// ---------------------------------------------------------------------------
extern "C" void kernel_launch(void* const* d_in, const int* in_sizes, int n_in,
                              void* d_out, int out_size, void* d_ws, size_t ws_size,
                              hipStream_t stream) {
    const float* nodes   = (const float*)d_in[0];   // (768,128)
    const float* adj     = (const float*)d_in[1];   // (768,768,1)
    const float* preW    = (const float*)d_in[2];   // (128,128)
    const float* postW   = (const float*)d_in[3];   // (512,128)
    const float* attn_w1 = (const float*)d_in[4];   // (256,256)
    const float* attn_b1 = (const float*)d_in[5];   // (256,)
    const float* attn_w2 = (const float*)d_in[6];   // (4,256)
    const float* attn_b2 = (const float*)d_in[7];   // (4,)
    float* out = (float*)d_out;                     // (768,512)

    // Workspace layout (floats)
    float* P  = (float*)d_ws;                       // 768*128
    float* Am = P  + (size_t)GN * GD;               // 768*256
    float* Bm = Am + (size_t)GN * GAW;              // 768*256
    float* V  = Bm + (size_t)GN * GAW;              // 768*512
    float* S  = V  + (size_t)GN * (GD * GH);        // 4*768*768

    constexpr int NT = 4;                           // 16x64 C strip per wave
    auto blocksFor = [](int M, int Nn) {
        const int tiles = (M >> 4) * (Nn / (16 * NT));
        return (tiles + 7) / 8;                     // 8 waves (256 thr) / block
    };

    // 1) P = nodes @ preW.T          (768x128)
    wmma_gemm_f32<true, NT><<<blocksFor(GN, GD), 256, 0, stream>>>(
        nodes, preW, P, GN, GD, GD, GD, 1, GD, GD, 1);
    // 2) Am = P @ W1a.T              W1a = attn_w1[:, :128]
    wmma_gemm_f32<true, NT><<<blocksFor(GN, GAW), 256, 0, stream>>>(
        P, attn_w1, Am, GN, GAW, GD, GD, 1, 2 * GD, GAW, 1);
    // 3) Bm = P @ W1b.T              W1b = attn_w1[:, 128:]
    wmma_gemm_f32<true, NT><<<blocksFor(GN, GAW), 256, 0, stream>>>(
        P, attn_w1 + GD, Bm, GN, GAW, GD, GD, 1, 2 * GD, GAW, 1);
    // 4) V = nodes @ postW.T         (768x512)
    wmma_gemm_f32<true, NT><<<blocksFor(GN, GD * GH), 256, 0, stream>>>(
        nodes, postW, V, GN, GD * GH, GD, GD, 1, GD, GD * GH, 1);

    // 5) pairwise logits -> planar S[h][i][j]
    attn_logits_kernel<<<dim3(GN / 16, GN / 16), 256, 0, stream>>>(
        Am, Bm, attn_b1, attn_w2, attn_b2, S);

    // 6) masked softmax over j, in place
    masked_softmax_kernel<<<dim3(GN, GH), 256, 0, stream>>>(S, adj);

    // 7) per head: out[:, f*4+h] = S[h] @ V_h, V_h[j,f] = V[j, f*4+h]
    for (int h = 0; h < GH; ++h) {
        wmma_gemm_f32<false, NT><<<blocksFor(GN, GD), 256, 0, stream>>>(
            S + (size_t)h * GN * GN,  // A: (768x768), lda=768
            V + h,                    // B(k,n) = V[k*512 + n*4 + h]
            out + h,                  // C[m,n] -> out[m*512 + n*4 + h]
            GN, GD, GN,
            GN, GD * GH, GH, GD * GH, GH);
    }
}